// FraudGraphSAGE_37761352466444
// MI455X (gfx1250) — compile-verified
//
#include <hip/hip_runtime.h>
#include <hip/hip_bf16.h>
#include <stdint.h>

typedef __attribute__((ext_vector_type(2))) float v2f;
typedef __attribute__((ext_vector_type(8))) float v8f;

// ---------------------------------------------------------------------------
// Zero a float region (grid-stride).
// ---------------------------------------------------------------------------
__global__ void zero_kernel(float* __restrict__ p, size_t n) {
    size_t i = (size_t)blockIdx.x * blockDim.x + threadIdx.x;
    size_t stride = (size_t)gridDim.x * blockDim.x;
    for (; i < n; i += stride) p[i] = 0.0f;
}

// ---------------------------------------------------------------------------
// Per-destination-node in-degree (float counts, for mean).
// ---------------------------------------------------------------------------
__global__ void count_kernel(const int* __restrict__ dst,
                             float* __restrict__ cnt, int E) {
    int t = blockIdx.x * blockDim.x + threadIdx.x;
    if (t < E) atomicAdd(&cnt[dst[t]], 1.0f);
}

// ---------------------------------------------------------------------------
// Edge scatter-add: agg[dst[e], :] += x[src[e], :].
// One thread handles one float4 chunk of one edge. dim = 4 << l2q.
// x / agg are L2-resident (51-102 MB vs 192 MB L2) so the random gather and
// the f32 atomics stay on-chip.
// ---------------------------------------------------------------------------
__global__ void scatter_kernel(const float* __restrict__ x,
                               const int* __restrict__ src,
                               const int* __restrict__ dst,
                               float* __restrict__ agg,
                               long long total, int l2q) {
    long long t = (long long)blockIdx.x * blockDim.x + threadIdx.x;
    if (t >= total) return;
    long long e = t >> l2q;
    int q = (int)(t & ((1 << l2q) - 1));
    int dim = 4 << l2q;
    size_t srow = (size_t)src[e] * (size_t)dim;
    size_t drow = (size_t)dst[e] * (size_t)dim;
    const float4 v = *(const float4*)(x + srow + (size_t)q * 4);
    float* o = agg + drow + (size_t)q * 4;
    atomicAdd(o + 0, v.x);
    atomicAdd(o + 1, v.y);
    atomicAdd(o + 2, v.z);
    atomicAdd(o + 3, v.w);
}

// ---------------------------------------------------------------------------
// Fused SAGEConv GEMM using V_WMMA_F32_16X16X4_F32 (fp32 in / fp32 acc):
//   out = relu( (agg/max(cnt,1)) @ Wl^T + bias + self @ Wr^T )
// One wave computes one 16x16 output tile; 8 waves/block cover 16 rows x 128
// cols; gridDim.y covers the remaining column blocks.
//
// Fragment layouts (ISA 7.12.2, wave32):
//   A (16x4 f32):  lane 0-15 -> M=lane, K={0,1}; lane 16-31 -> M=lane-16, K={2,3}
//   B (4x16 f32):  lane 0-15 -> N=lane, K={0,1}; lane 16-31 -> N=lane-16, K={2,3}
//   C/D (16x16):   VGPR r: lanes 0-15 M=r, lanes 16-31 M=8+r; N = lane&15
// ---------------------------------------------------------------------------
__global__ void __launch_bounds__(256) sage_gemm_kernel(
    const float* __restrict__ aggA,   // [N,K] neighbor sums
    const float* __restrict__ selfA,  // [N,K] self features
    const float* __restrict__ cnt,    // [N]
    const float* __restrict__ Wl,     // [NC,K] row-major
    const float* __restrict__ Wr,     // [NC,K] row-major
    const float* __restrict__ bias,   // [NC]
    float* __restrict__ out,          // [N,NC]
    int N, int K, int NC) {
    const int lane = threadIdx.x & 31;
    const int wave = threadIdx.x >> 5;
    const int row0 = blockIdx.x * 16;
    const int col0 = blockIdx.y * 128 + wave * 16;
    if (row0 >= N || col0 >= NC) return;  // block/wave-uniform: EXEC all-ones

    const int mn = lane & 15;   // M for A-side, N for B/C/D-side
    const int kp = lane >> 4;   // which K pair (0 -> K=0,1 ; 1 -> K=2,3)

    const float c = cnt[row0 + mn];
    const float invc = 1.0f / fmaxf(c, 1.0f);

    const float* aRow  = aggA  + (size_t)(row0 + mn) * K;
    const float* sRow  = selfA + (size_t)(row0 + mn) * K;
    const float* wlRow = Wl    + (size_t)(col0 + mn) * K;
    const float* wrRow = Wr    + (size_t)(col0 + mn) * K;

    v8f acc = {};
    for (int k0 = 0; k0 < K; k0 += 4) {
        const int kk = k0 + 2 * kp;
        // lin_l path: mean neighbors
        float2 av = *(const float2*)(aRow + kk);
        float2 bv = *(const float2*)(wlRow + kk);
        v2f a1; a1.x = av.x * invc; a1.y = av.y * invc;
        v2f b1; b1.x = bv.x;        b1.y = bv.y;
        acc = __builtin_amdgcn_wmma_f32_16x16x4_f32(
            false, a1, false, b1, (short)0, acc, false, false);
        // lin_r path: self features
        float2 sv = *(const float2*)(sRow + kk);
        float2 wv = *(const float2*)(wrRow + kk);
        v2f a2; a2.x = sv.x; a2.y = sv.y;
        v2f b2; b2.x = wv.x; b2.y = wv.y;
        acc = __builtin_amdgcn_wmma_f32_16x16x4_f32(
            false, a2, false, b2, (short)0, acc, false, false);
    }

    const float badd = bias[col0 + mn];
    const int mbase = 8 * kp;
#pragma unroll
    for (int r = 0; r < 8; ++r) {
        float v = acc[r] + badd;
        v = fmaxf(v, 0.0f);  // relu
        out[(size_t)(row0 + mbase + r) * NC + (col0 + mn)] = v;
    }
}

// ---------------------------------------------------------------------------
// Output head: logits = h @ W_out^T + b_out (2 classes), softmax, emit p[:,1].
// One wave per node; lanes stride over K with shuffle reduction (wave32).
// ---------------------------------------------------------------------------
__global__ void logits_kernel(const float* __restrict__ h,
                              const float* __restrict__ Wout,  // [2,K]
                              const float* __restrict__ bout,  // [2]
                              float* __restrict__ out, int N, int K) {
    int node = blockIdx.x * (blockDim.x >> 5) + (threadIdx.x >> 5);
    int lane = threadIdx.x & 31;
    if (node >= N) return;
    const float* row = h + (size_t)node * K;
    float s0 = 0.0f, s1 = 0.0f;
    for (int k = lane; k < K; k += 32) {
        float v = row[k];
        s0 += v * Wout[k];
        s1 += v * Wout[K + k];
    }
#pragma unroll
    for (int off = 16; off > 0; off >>= 1) {
        s0 += __shfl_down(s0, off, 32);
        s1 += __shfl_down(s1, off, 32);
    }
    if (lane == 0) {
        float l0 = s0 + bout[0];
        float l1 = s1 + bout[1];
        float m = fmaxf(l0, l1);
        float e0 = __expf(l0 - m);
        float e1 = __expf(l1 - m);
        out[node] = e1 / (e0 + e1);
    }
}

// ---------------------------------------------------------------------------
// Launcher
// ---------------------------------------------------------------------------
extern "C" void kernel_launch(void* const* d_in, const int* in_sizes, int n_in,
                              void* d_out, int out_size, void* d_ws, size_t ws_size,
                              hipStream_t stream) {
    const float* x    = (const float*)d_in[0];  // [N,128]
    const int*   ei   = (const int*)d_in[1];    // [2,E] (int32: JAX x64-off)
    const float* W1l  = (const float*)d_in[2];  // [256,128]
    const float* b1l  = (const float*)d_in[3];  // [256]
    const float* W1r  = (const float*)d_in[4];  // [256,128]
    const float* W2l  = (const float*)d_in[5];  // [256,256]
    const float* b2l  = (const float*)d_in[6];  // [256]
    const float* W2r  = (const float*)d_in[7];  // [256,256]
    const float* Wout = (const float*)d_in[8];  // [2,256]
    const float* bout = (const float*)d_in[9];  // [2]
    float* out = (float*)d_out;

    const int N = in_sizes[0] / 128;
    const int E = in_sizes[1] / 2;
    const int* src = ei;
    const int* dst = ei + E;

    // Workspace carve-up (floats): cnt[N] | h1[N*256] | agg[N*256] | h2[N*256]
    float* wsf = (float*)d_ws;
    float* cnt = wsf;
    float* h1  = cnt + N;
    float* agg = h1 + (size_t)N * 256;
    float* h2  = agg + (size_t)N * 256;

    const int TB = 256;

    // ---- Layer 1 ----
    zero_kernel<<<1024, TB, 0, stream>>>(cnt, (size_t)N);
    zero_kernel<<<4096, TB, 0, stream>>>(agg, (size_t)N * 128);
    count_kernel<<<(E + TB - 1) / TB, TB, 0, stream>>>(dst, cnt, E);
    {
        long long total = (long long)E * 32;  // 128 feats / 4 per thread
        scatter_kernel<<<(int)((total + TB - 1) / TB), TB, 0, stream>>>(
            x, src, dst, agg, total, 5);
    }
    {
        dim3 g((N + 15) / 16, 2);  // 2 col-blocks of 128 => NC=256
        sage_gemm_kernel<<<g, TB, 0, stream>>>(agg, x, cnt, W1l, W1r, b1l, h1,
                                               N, 128, 256);
    }

    // ---- Layer 2 ----
    zero_kernel<<<8192, TB, 0, stream>>>(agg, (size_t)N * 256);
    {
        long long total = (long long)E * 64;  // 256 feats / 4 per thread
        scatter_kernel<<<(int)((total + TB - 1) / TB), TB, 0, stream>>>(
            h1, src, dst, agg, total, 6);
    }
    {
        dim3 g((N + 15) / 16, 2);
        sage_gemm_kernel<<<g, TB, 0, stream>>>(agg, h1, cnt, W2l, W2r, b2l, h2,
                                               N, 256, 256);
    }

    // ---- Output head ----
    logits_kernel<<<(N + 7) / 8, TB, 0, stream>>>(h2, Wout, bout, out, N, 256);
}